// Kernel_Correlation_80985903334294
// MI455X (gfx1250) — compile-verified
//
#include <hip/hip_runtime.h>
#include <hip/hip_bf16.h>

typedef __attribute__((ext_vector_type(2))) float v2f;
typedef __attribute__((ext_vector_type(8))) float v8f;

#define NKER 64
#define KSUB 16
#define WAVES 8
#define PTS_PER_WAVE 16
#define PTS_PER_BLOCK (WAVES * PTS_PER_WAVE)   // 128

// out[i][m] = (1/128) * sum_l exp(-(||n_i||^2 + ||k_{m,l}||^2 - 2 n_i.k_{m,l}))
__global__ __launch_bounds__(256)
void Kernel_Correlation_80985903334294_kernel(const float* __restrict__ normal,
                                              const float* __restrict__ kern,
                                              float* __restrict__ out)
{
    // LDS: A-fragments per kernel m (16 KB), -0.5*||k||^2 (4 KB), staging (34 KB)
    __shared__ __align__(16) v2f   sA[NKER * 32];
    __shared__ __align__(16) float sK2[NKER * KSUB];
    __shared__ __align__(16) float sOut[WAVES][16 * 68];  // pad 68: conflict-free, 16B-aligned

    const int tid  = threadIdx.x;
    const int lane = tid & 31;
    const int wave = tid >> 5;
    const int col  = lane & 15;
    const int hi   = lane >> 4;              // 0: rows 0-7 / K=0,1 ; 1: rows 8-15 / K=2,3

    float* myOut = sOut[wave];

    // ---- zero this wave's accumulation staging (atomicAdd target) ----
    {
        float4 z; z.x = 0.f; z.y = 0.f; z.z = 0.f; z.w = 0.f;
        for (int t = lane; t < (16 * 68) / 4; t += 32)
            ((float4*)myOut)[t] = z;
    }

    // ---- cooperative preload of the 1024 kernel points into WMMA A-layout ----
    for (int p = tid; p < NKER * KSUB; p += 256) {
        float kx = kern[p * 3 + 0];
        float ky = kern[p * 3 + 1];
        float kz = kern[p * 3 + 2];
        sK2[p] = -0.5f * (kx * kx + ky * ky + kz * kz);
        int m = p >> 4, l = p & 15;
        v2f lo;  lo.x  = kx; lo.y  = ky;     // (M=l, K=0|1)
        v2f hi2; hi2.x = kz; hi2.y = 0.0f;   // (M=l, K=2|3)
        sA[m * 32 + l]      = lo;
        sA[m * 32 + l + 16] = hi2;
    }
    __syncthreads();

    // ---- per-wave B fragment: 16 points as 4x16 f32 ----
    const int base = blockIdx.x * PTS_PER_BLOCK + wave * PTS_PER_WAVE;
    const int i    = base + col;
    const float nx = normal[i * 3 + 0];
    const float ny = normal[i * 3 + 1];
    const float nz = normal[i * 3 + 2];
    v2f b;
    b.x = hi ? nz : nx;                      // K=0 or K=2
    b.y = hi ? 0.0f : ny;                    // K=1 or K=3
    const float LOG2E = 1.4426950408889634f;
    const float S     = 2.0f * LOG2E;        // exp(2d) = exp2(d*S)
    // bias folds -||n||^2 AND the 1/128 scale: exp(x)/128 = exp2(x*log2e - 7)
    const float nb    = __builtin_fmaf(-(nx * nx + ny * ny + nz * nz), LOG2E, -7.0f);

    // ---- software-pipelined main loop over the 64 kernels ----
    v2f    a   = sA[lane];                   // m = 0 fragment
    float4 k2a = *(const float4*)&sK2[hi * 8];
    float4 k2b = *(const float4*)&sK2[hi * 8 + 4];

    for (int m = 0; m < NKER; ++m) {
        v8f c;
        c[0] = k2a.x; c[1] = k2a.y; c[2] = k2a.z; c[3] = k2a.w;
        c[4] = k2b.x; c[5] = k2b.y; c[6] = k2b.z; c[7] = k2b.w;

        // D = A x B + C  ->  d[v] = dot - 0.5*||k||^2
        v8f d = __builtin_amdgcn_wmma_f32_16x16x4_f32(
            /*neg_a=*/false, a, /*neg_b=*/false, b,
            /*c_mod=*/(short)0, c, /*reuse_a=*/false, /*reuse_b=*/false);

        // prefetch next iteration's fragments while the exp chain runs
        if (m + 1 < NKER) {
            a   = sA[(m + 1) * 32 + lane];
            k2a = *(const float4*)&sK2[(m + 1) * 16 + hi * 8];
            k2b = *(const float4*)&sK2[(m + 1) * 16 + hi * 8 + 4];
        }

        // exp2(fma(d, S, nb)) = exp(-d^2)/128; always finite (arg <= -7)
        float s = __builtin_amdgcn_exp2f(__builtin_fmaf(d[0], S, nb))
                + __builtin_amdgcn_exp2f(__builtin_fmaf(d[1], S, nb))
                + __builtin_amdgcn_exp2f(__builtin_fmaf(d[2], S, nb))
                + __builtin_amdgcn_exp2f(__builtin_fmaf(d[3], S, nb))
                + __builtin_amdgcn_exp2f(__builtin_fmaf(d[4], S, nb))
                + __builtin_amdgcn_exp2f(__builtin_fmaf(d[5], S, nb))
                + __builtin_amdgcn_exp2f(__builtin_fmaf(d[6], S, nb))
                + __builtin_amdgcn_exp2f(__builtin_fmaf(d[7], S, nb));

        // accumulate both lane-halves (rows 0-7 and 8-15) without shuffle/wait:
        // ds_add_f32 (no return) -- lanes c and c^16 serialize on one address
        atomicAdd(&myOut[col * 68 + m], s);
    }

    __syncthreads();   // drain DS atomics before reading staging

    // ---- coalesced float4 writeback: 16 points x 64 kernels per wave ----
    for (int j = 0; j < 8; ++j) {
        int q  = j * 32 + lane;              // 0..255 -> float4 index in wave tile
        int c2 = q >> 4;                     // point within wave
        int m0 = (q & 15) << 2;              // kernel group of 4
        float4 v = *(const float4*)&myOut[c2 * 68 + m0];
        *(float4*)&out[(size_t)base * 64 + (size_t)q * 4] = v;
    }
}

extern "C" void kernel_launch(void* const* d_in, const int* in_sizes, int n_in,
                              void* d_out, int out_size, void* d_ws, size_t ws_size,
                              hipStream_t stream) {
    const float* normal = (const float*)d_in[0];
    // d_in[1] = neighbour indices: gathered but never used by the reference math
    const float* kern   = (const float*)d_in[2];
    float* out = (float*)d_out;

    const int n = in_sizes[0] / 3;                 // 262144 points
    dim3 grid(n / PTS_PER_BLOCK);                  // 2048 blocks
    dim3 block(256);                               // 8 waves (wave32)
    Kernel_Correlation_80985903334294_kernel<<<grid, block, 0, stream>>>(normal, kern, out);
}